// FastMoEGCN_44178033607221
// MI455X (gfx1250) — compile-verified
//
#include <hip/hip_runtime.h>
#include <hip/hip_bf16.h>

// Problem constants (from reference)
constexpr int Bc = 64;      // graphs
constexpr int Nn = 512;     // nodes
constexpr int Hc = 1024;    // hidden
constexpr int Ec = 4;       // experts
constexpr float BN_EPS = 1e-5f;

// GEMM tiling: block tile 64x128, 8 waves as 2x4, each wave a 32x32 tile
// (2x2 register blocking of 16x16 WMMA accumulators).
constexpr int TM  = 64;     // block tile rows (M)
constexpr int TN  = 128;    // block tile cols (N)
constexpr int TKB = 32;     // K block

typedef __attribute__((ext_vector_type(2))) float v2f;
typedef __attribute__((ext_vector_type(8))) float v8f;

// ---------------------------------------------------------------------------
// Block GEMM: C[m0:m0+64, n0:n0+128] = A[M,K] * B[K,N]  (row-major, fp32)
// 256 threads = 8 waves; each wave owns four 16x16 WMMA accumulators (32x32).
// Uses V_WMMA_F32_16X16X4_F32 (CDNA5 fp32 matrix pipe).
// 2x2 register blocking -> 1 DS instruction per WMMA instead of 2.
// ---------------------------------------------------------------------------
__device__ __forceinline__ void gemm_block_f32(
    const float* __restrict__ A, int lda,
    const float* __restrict__ Bm, int ldb,
    float* __restrict__ C, int ldc,
    int K, int m0, int n0)
{
    __shared__ float As[TM][TKB + 1];   // 64 x 33 (+1 pad: conflict-free frag reads)
    __shared__ float Bs[TKB][TN];       // 32 x 128

    const int tid   = threadIdx.x;
    const int lane  = tid & 31;
    const int wave  = tid >> 5;         // 0..7
    const int wm    = wave & 1;         // 2 M-groups of 32
    const int wn    = wave >> 1;        // 4 N-groups of 32
    const int lhalf = lane >> 4;        // 0/1
    const int l15   = lane & 15;
    const int kb    = lhalf * 2;        // K offset per half-wave (ISA 16x4 layout)

    v8f acc00 = {}, acc01 = {}, acc10 = {}, acc11 = {};

    const int ma0 = wm * 32 + l15;          // A rows for the two M-subtiles
    const int ma1 = ma0 + 16;
    const int nb0 = wn * 32 + l15;          // B cols for the two N-subtiles
    const int nb1 = nb0 + 16;

    for (int k0 = 0; k0 < K; k0 += TKB) {
        // --- stage A tile (64x32): two float4 per thread ---
        #pragma unroll
        for (int r = 0; r < 2; ++r) {
            const int idx = tid + r * 256;
            const int row = idx >> 3;               // 0..63
            const int c4  = (idx & 7) << 2;         // 0..28
            const float4 va = *(const float4*)(A + (size_t)(m0 + row) * lda + k0 + c4);
            As[row][c4 + 0] = va.x; As[row][c4 + 1] = va.y;
            As[row][c4 + 2] = va.z; As[row][c4 + 3] = va.w;
        }
        // --- stage B tile (32x128): four float4 per thread ---
        #pragma unroll
        for (int r = 0; r < 4; ++r) {
            const int idx = tid + r * 256;
            const int row = idx >> 5;               // 0..31
            const int c4  = (idx & 31) << 2;        // 0..124
            *(float4*)(&Bs[row][c4]) =
                *(const float4*)(Bm + (size_t)(k0 + row) * ldb + n0 + c4);
        }
        // uniform-branch prefetch of next B K-slab (global_prefetch_b8)
        if (k0 + TKB < K) {
            __builtin_prefetch(Bm + (size_t)(k0 + TKB) * ldb + n0 + ((tid & 31) << 2), 0, 0);
        }
        __syncthreads();

        #pragma unroll
        for (int kk = 0; kk < TKB; kk += 4) {
            v2f a0, a1, b0, b1;
            a0[0] = As[ma0][kk + kb];
            a0[1] = As[ma0][kk + kb + 1];
            a1[0] = As[ma1][kk + kb];
            a1[1] = As[ma1][kk + kb + 1];
            b0[0] = Bs[kk + kb][nb0];
            b0[1] = Bs[kk + kb + 1][nb0];
            b1[0] = Bs[kk + kb][nb1];
            b1[1] = Bs[kk + kb + 1][nb1];
            acc00 = __builtin_amdgcn_wmma_f32_16x16x4_f32(false, a0, false, b0, (short)0, acc00, false, false);
            acc01 = __builtin_amdgcn_wmma_f32_16x16x4_f32(false, a0, false, b1, (short)0, acc01, false, false);
            acc10 = __builtin_amdgcn_wmma_f32_16x16x4_f32(false, a1, false, b0, (short)0, acc10, false, false);
            acc11 = __builtin_amdgcn_wmma_f32_16x16x4_f32(false, a1, false, b1, (short)0, acc11, false, false);
        }
        __syncthreads();
    }

    // --- store C per ISA 16x16 f32 C/D layout ---
    const int ncol0 = n0 + wn * 32 + l15;
    const int ncol1 = ncol0 + 16;
    const int mb0   = m0 + wm * 32 + lhalf * 8;
    const int mb1   = mb0 + 16;
    #pragma unroll
    for (int i = 0; i < 8; ++i) {
        C[(size_t)(mb0 + i) * ldc + ncol0] = acc00[i];
        C[(size_t)(mb0 + i) * ldc + ncol1] = acc01[i];
        C[(size_t)(mb1 + i) * ldc + ncol0] = acc10[i];
        C[(size_t)(mb1 + i) * ldc + ncol1] = acc11[i];
    }
}

// ---------------------------------------------------------------------------
// Router: scores = mean_n(x[b]) @ Rw + Rb ; top1 = argmax (first max wins)
// ---------------------------------------------------------------------------
__global__ __launch_bounds__(256) void router_k(
    const float* __restrict__ x, const float* __restrict__ rw,
    const float* __restrict__ rb, int* __restrict__ top1)
{
    __shared__ float meanv[Hc];
    __shared__ float sc[Ec];
    const int b = blockIdx.x;
    const int t = threadIdx.x;

    #pragma unroll
    for (int j = 0; j < Hc / 256; ++j) {
        const int h = t + j * 256;
        const float* p = x + (size_t)b * Nn * Hc + h;
        float s = 0.f;
        for (int n = 0; n < Nn; ++n) s += p[(size_t)n * Hc];
        meanv[h] = s * (1.0f / Nn);
    }
    __syncthreads();
    if (t < Ec) {
        float s = rb[t];
        for (int h = 0; h < Hc; ++h) s += meanv[h] * rw[h * Ec + t];
        sc[t] = s;
    }
    __syncthreads();
    if (t == 0) {
        int best = 0; float bv = sc[0];
        for (int e = 1; e < Ec; ++e) if (sc[e] > bv) { bv = sc[e]; best = e; }
        top1[b] = best;
    }
}

// ---------------------------------------------------------------------------
// support[b] = x[b] (512x1024) @ expert_w[top1[b]] (1024x1024)
// ---------------------------------------------------------------------------
__global__ __launch_bounds__(256) void gemm_support_k(
    const float* __restrict__ x, const float* __restrict__ ew,
    const int* __restrict__ top1, float* __restrict__ support)
{
    const int b = blockIdx.z;
    const int e = top1[b];                       // uniform per block
    gemm_block_f32(x + (size_t)b * Nn * Hc, Hc,
                   ew + (size_t)e * Hc * Hc, Hc,
                   support + (size_t)b * Nn * Hc, Hc,
                   Hc, blockIdx.y * TM, blockIdx.x * TN);
}

// ---------------------------------------------------------------------------
// out[b] = adj[b] (512x512) @ support[b] (512x1024)   -> written to d_out
// ---------------------------------------------------------------------------
__global__ __launch_bounds__(256) void gemm_out_k(
    const float* __restrict__ adj, const float* __restrict__ support,
    float* __restrict__ out)
{
    const int b = blockIdx.z;
    gemm_block_f32(adj + (size_t)b * Nn * Nn, Nn,
                   support + (size_t)b * Nn * Hc, Hc,
                   out + (size_t)b * Nn * Hc, Hc,
                   Nn, blockIdx.y * TM, blockIdx.x * TN);
}

// ---------------------------------------------------------------------------
// BN stage 1: per (b,h) sum / sumsq over the N axis (no atomics)
// ---------------------------------------------------------------------------
__global__ __launch_bounds__(256) void bn_partial_k(
    const float* __restrict__ out,
    float* __restrict__ psum, float* __restrict__ psumsq)
{
    const int h = blockIdx.x * 256 + threadIdx.x;
    const int b = blockIdx.y;
    const float* p = out + (size_t)b * Nn * Hc + h;
    float s = 0.f, q = 0.f;
    for (int n = 0; n < Nn; ++n) { const float v = p[(size_t)n * Hc]; s += v; q += v * v; }
    psum[b * Hc + h]   = s;
    psumsq[b * Hc + h] = q;
}

// ---------------------------------------------------------------------------
// BN stage 2: per (e,h) deterministic reduce over graphs; fold BN into
// scale/shift:  y = out*scale + shift
// ---------------------------------------------------------------------------
__global__ __launch_bounds__(256) void bn_scaleshift_k(
    const float* __restrict__ psum, const float* __restrict__ psumsq,
    const int* __restrict__ top1,
    const float* __restrict__ gamma, const float* __restrict__ beta,
    float* __restrict__ scale, float* __restrict__ shift)
{
    const int i = blockIdx.x * 256 + threadIdx.x;   // e*H + h
    if (i >= Ec * Hc) return;
    const int e = i >> 10;
    const int h = i & (Hc - 1);
    float s = 0.f, q = 0.f; int c = 0;
    for (int b = 0; b < Bc; ++b) {
        if (top1[b] == e) { s += psum[b * Hc + h]; q += psumsq[b * Hc + h]; ++c; }
    }
    const float cnt  = fmaxf((float)c * (float)Nn, 1.0f);
    const float mean = s / cnt;
    const float var  = q / cnt - mean * mean;
    const float inv  = rsqrtf(var + BN_EPS);
    const float sc   = gamma[i] * inv;
    scale[i] = sc;
    shift[i] = beta[i] - mean * sc;
}

// ---------------------------------------------------------------------------
// BN stage 3: in-place affine + ReLU on d_out
// ---------------------------------------------------------------------------
__global__ __launch_bounds__(256) void bn_finalize_k(
    float* __restrict__ out, const int* __restrict__ top1,
    const float* __restrict__ scale, const float* __restrict__ shift)
{
    const int b   = blockIdx.y;
    const int pos = blockIdx.x * 256 + threadIdx.x;     // 0 .. N*H-1
    const int h   = pos & (Hc - 1);
    const int e   = top1[b];
    const size_t idx = (size_t)b * Nn * Hc + pos;
    const float y = out[idx] * scale[e * Hc + h] + shift[e * Hc + h];
    out[idx] = fmaxf(y, 0.0f);
}

// ---------------------------------------------------------------------------
// Launcher
// ---------------------------------------------------------------------------
extern "C" void kernel_launch(void* const* d_in, const int* in_sizes, int n_in,
                              void* d_out, int out_size, void* d_ws, size_t ws_size,
                              hipStream_t stream) {
    const float* x     = (const float*)d_in[0];   // [B,N,H]
    const float* adj   = (const float*)d_in[1];   // [B,N,N]
    const float* rw    = (const float*)d_in[2];   // [H,E]
    const float* rb    = (const float*)d_in[3];   // [E]
    const float* ew    = (const float*)d_in[4];   // [E,H,H]
    const float* gamma = (const float*)d_in[5];   // [E,H]
    const float* beta  = (const float*)d_in[6];   // [E,H]
    float* out = (float*)d_out;                   // [B,N,H]

    // Workspace layout (256B-aligned offsets)
    char* wsb = (char*)d_ws;
    int*   top1    = (int*)(wsb + 0);                              //  64 ints
    float* psum    = (float*)(wsb + 1024);                         //  B*H floats (256KB)
    float* psumsq  = (float*)(wsb + 1024 + 262144);                //  B*H floats
    float* scale   = (float*)(wsb + 1024 + 2 * 262144);            //  E*H floats
    float* shift   = (float*)(wsb + 1024 + 2 * 262144 + 16384);    //  E*H floats
    float* support = (float*)(wsb + (size_t)1024 * 1024);          //  B*N*H floats (128MB)

    // 1) routing
    router_k<<<dim3(Bc), dim3(256), 0, stream>>>(x, rw, rb, top1);

    // 2) support = x @ W[top1]   (WMMA f32)
    gemm_support_k<<<dim3(Hc / TN, Nn / TM, Bc), dim3(256), 0, stream>>>(x, ew, top1, support);

    // 3) out = adj @ support     (WMMA f32)
    gemm_out_k<<<dim3(Hc / TN, Nn / TM, Bc), dim3(256), 0, stream>>>(adj, support, out);

    // 4) BN stats (deterministic, two-stage)
    bn_partial_k<<<dim3(Hc / 256, Bc), dim3(256), 0, stream>>>(out, psum, psumsq);
    bn_scaleshift_k<<<dim3((Ec * Hc + 255) / 256), dim3(256), 0, stream>>>(
        psum, psumsq, top1, gamma, beta, scale, shift);

    // 5) affine + ReLU in place
    bn_finalize_k<<<dim3((Nn * Hc) / 256, Bc), dim3(256), 0, stream>>>(out, top1, scale, shift);
}